// ROIRotate_87351044866639
// MI455X (gfx1250) — compile-verified
//
#include <hip/hip_runtime.h>

// ROI crop + bilinear resample (ROIRotate with angle==0), MI455X / gfx1250.
// Bandwidth-bound: ~128 MB NT stores + L2-resident gathers from a 16 MB map.
// No matmul structure -> wmma not applicable; optimize the memory path instead.

typedef float v4f __attribute__((ext_vector_type(4)));

#define ROI_HEIGHT 8
#define C_CH       32
#define C_VEC      (C_CH / 4)   // float4 groups per channel dim = 8
#define FM_H       128
#define FM_W       128

__device__ __forceinline__ int clampi(int x, int lo, int hi) {
    x = (x < lo) ? lo : x;
    return (x > hi) ? hi : x;
}

// grid: (ceil(MW*C_VEC/256), B*ROI_HEIGHT), block: 256 threads (8 waves).
// blockIdx.y encodes (b, v); threads in x cover (u, channel-vec).
// Lanes 8k..8k+7 of a wave fetch one contiguous 128-B feature row each gather.
__global__ __launch_bounds__(256)
void roi_crops_kernel(const float* __restrict__ fm,
                      const float* __restrict__ boxes,
                      const int*   __restrict__ box_idx,
                      float*       __restrict__ out,
                      int MW)
{
    const int rowVec = MW * C_VEC;                       // float4 elems per (b,v) row
    int idx = blockIdx.x * blockDim.x + threadIdx.x;     // within the (b,v) row
    if (idx >= rowVec) return;

    const int cv = idx & (C_VEC - 1);                    // float4 index in channels
    const int u  = idx >> 3;                             // C_VEC == 8
    const int bv = blockIdx.y;
    const int v  = bv & (ROI_HEIGHT - 1);                // HEIGHT == 8
    const int b  = bv >> 3;

    // Box parameters (scaled by 1/4 as in reference).
    const float* bx = boxes + (long)b * 9;
    const float x1 = bx[0] * 0.25f, y1 = bx[1] * 0.25f;
    const float x2 = bx[2] * 0.25f, y2 = bx[3] * 0.25f;
    const float x4 = bx[6] * 0.25f, y4 = bx[7] * 0.25f;

    const float box_w = x2 - x1;
    const float box_h = y4 - y1;
    float width = (float)ROI_HEIGHT * box_w / fmaxf(box_h, 1e-6f);
    width = fminf(fmaxf(width, 1.0f), (float)MW);

    const float uf = (float)u;
    const float tu = uf / width;                          // exact f32 division, as reference
    const float tv = (float)v * (1.0f / (float)ROI_HEIGHT);

    const float sx = x1 + tu * (x2 - x1) + tv * (x4 - x1);
    const float sy = y1 + tu * (y2 - y1) + tv * (y4 - y1);

    const float fx0 = floorf(sx), fy0 = floorf(sy);
    const float wx = sx - fx0, wy = sy - fy0;

    const int x0i = clampi((int)fx0, 0, FM_W - 1);
    const int x1i = (x0i + 1 > FM_W - 1) ? (FM_W - 1) : (x0i + 1);
    const int y0i = clampi((int)fy0, 0, FM_H - 1);
    const int y1i = (y0i + 1 > FM_H - 1) ? (FM_H - 1) : (y0i + 1);

    const int bi = box_idx[b];

    const v4f* __restrict__ fm4 = (const v4f*)fm;
    const long imgBase = (long)bi * FM_H;
    const long o00 = ((imgBase + y0i) * FM_W + x0i) * C_VEC + cv;
    const long o01 = ((imgBase + y0i) * FM_W + x1i) * C_VEC + cv;
    const long o10 = ((imgBase + y1i) * FM_W + x0i) * C_VEC + cv;
    const long o11 = ((imgBase + y1i) * FM_W + x1i) * C_VEC + cv;

    // Four global_load_b128 gathers; RT temporal hint keeps the 16 MB map in L2.
    const v4f g00 = fm4[o00];
    const v4f g01 = fm4[o01];
    const v4f g10 = fm4[o10];
    const v4f g11 = fm4[o11];

    const float owx = 1.0f - wx, owy = 1.0f - wy;
    v4f top = owx * g00 + wx * g01;                       // same blend form as reference
    v4f bot = owx * g10 + wx * g11;
    v4f val = owy * top + wy * bot;

    if (!(uf < width)) {                                  // float compare, as reference
        v4f zero = {0.0f, 0.0f, 0.0f, 0.0f};
        val = zero;
    }

    // Streaming 128 MB output: non-temporal so it doesn't evict the feature map.
    v4f* __restrict__ out4 = (v4f*)out;
    const long off = (long)bv * rowVec + idx;
    __builtin_nontemporal_store(val, out4 + off);
}

__global__ __launch_bounds__(256)
void roi_padded_width_kernel(const float* __restrict__ boxes,
                             float* __restrict__ out_pw,
                             int B, int MW)
{
    int b = blockIdx.x * blockDim.x + threadIdx.x;
    if (b >= B) return;
    const float* bx = boxes + (long)b * 9;
    const float x1 = bx[0] * 0.25f, y1 = bx[1] * 0.25f;
    const float x2 = bx[2] * 0.25f, y4 = bx[7] * 0.25f;
    float width = (float)ROI_HEIGHT * (x2 - x1) / fmaxf(y4 - y1, 1e-6f);
    width = fminf(fmaxf(width, 1.0f), (float)MW);
    __builtin_nontemporal_store((float)MW - width, out_pw + b);
}

extern "C" void kernel_launch(void* const* d_in, const int* in_sizes, int n_in,
                              void* d_out, int out_size, void* d_ws, size_t ws_size,
                              hipStream_t stream) {
    const float* fm    = (const float*)d_in[0];   // (N,128,128,32) f32
    const float* boxes = (const float*)d_in[1];   // (B,9) f32
    const int*   bidx  = (const int*)d_in[2];     // (B,) int
    // d_in[3] is max_width on device; derive it from sizes instead (no readback,
    // keeps the launch graph-capture safe & deterministic).

    const int B = in_sizes[1] / 9;                            // 2048
    const long per = (long)out_size / B;                      // HEIGHT*MW*C + 1
    const int MW = (int)((per - 1) / (ROI_HEIGHT * C_CH));    // 64
    const long cropElems = (long)B * ROI_HEIGHT * MW * C_CH;  // 33,554,432

    float* out = (float*)d_out;

    const int rowVec = MW * C_VEC;                            // 512 float4 per (b,v)
    dim3 block(256);
    dim3 grid((rowVec + 255) / 256, (unsigned)(B * ROI_HEIGHT));
    roi_crops_kernel<<<grid, block, 0, stream>>>(fm, boxes, bidx, out, MW);

    roi_padded_width_kernel<<<dim3((B + 255) / 256), dim3(256), 0, stream>>>(
        boxes, out + cropElems, B, MW);
}